// HistogramObserver_17987323036277
// MI455X (gfx1250) — compile-verified
//
#include <hip/hip_runtime.h>
#include <math.h>

// ---------------- HistogramObserver for MI455X (gfx1250, wave32) ----------------
// Bandwidth-bound: two passes over 256MB of f32 (min/max, then histc).
// Floor ~= 512MB / 23.3TB/s ~= 22us; everything else is O(2048).
// Pass 1 uses GLOBAL_LOAD_ASYNC_TO_LDS_B128 (ASYNCcnt DMA) in an 8-deep ring.
// Pass 2 keeps direct b128 loads (LDS bandwidth is reserved for ds_add_u32).

#define BINS   2048
#define UPR    128
#define TPB    256
#define NB_MM  1024          // blocks for min/max pass
#define NB_H   1024          // blocks for histogram pass
#define NC     4             // LDS histogram copies (contention reduction)
#define CSTRIDE 2064         // 2048 + 16 pad -> copies land in different bank groups
#define DEPTH  8             // async staging ring depth (s_wait_asynccnt 0x7)

// workspace layout (float offsets within d_ws)
#define OFF_PMIN   0                    // [NB_MM] per-block min
#define OFF_PMAX   NB_MM                // [NB_MM] per-block max
#define OFF_SCAL   (2*NB_MM)            // f[0]=cmin f[1]=bw ; i[2]=start_idx i[3]=dr
#define OFF_COUNTS (OFF_SCAL + 16)      // [BINS] u32 counts
#define OFF_C      (OFF_COUNTS + BINS)  // [BINS+1] cumsum of old histogram
// total floats used: OFF_C + BINS + 1 = 6161  (~24.6 KB of d_ws)

__device__ __forceinline__ float wave_min(float v) {
#pragma unroll
  for (int m = 16; m > 0; m >>= 1) v = fminf(v, __shfl_xor(v, m, 32));
  return v;
}
__device__ __forceinline__ float wave_max(float v) {
#pragma unroll
  for (int m = 16; m > 0; m >>= 1) v = fmaxf(v, __shfl_xor(v, m, 32));
  return v;
}

// async DMA: global (64-bit vaddr pair) -> LDS (32-bit wave-relative byte addr)
__device__ __forceinline__ void async_copy_b128(unsigned lds_byte_addr, const float4* gaddr) {
  asm volatile("global_load_async_to_lds_b128 %0, %1, off"
               :: "v"(lds_byte_addr), "v"(gaddr) : "memory");
}

// ---- kernel 0: zero the global count accumulators (graph replays need this) ----
__global__ void hobs_init_kernel(unsigned* __restrict__ counts) {
  int i = blockIdx.x * blockDim.x + threadIdx.x;
  if (i < BINS) counts[i] = 0u;
}

// ---- kernel 1: async-LDS staged grid-stride min/max -> per-block partials ----
__global__ void hobs_minmax_kernel(const float* __restrict__ x, int n,
                                   float* __restrict__ pmin, float* __restrict__ pmax) {
  __shared__ float4 stage[DEPTH * TPB];   // 32KB ring: lane i owns stage[slot*TPB+i]
  float lmin = INFINITY, lmax = -INFINITY;
  const float4* __restrict__ x4 = (const float4*)x;
  const int n4 = n >> 2;
  const int threads = gridDim.x * blockDim.x;
  const int tid = blockIdx.x * blockDim.x + threadIdx.x;
  const int tx = threadIdx.x;
  const int iters = n4 / threads;         // uniform tile count (no per-lane guards)
  int rem_start = 0;

  if (iters > DEPTH) {
    rem_start = iters * threads;
    // prologue: fill the ring
#pragma unroll
    for (int d = 0; d < DEPTH; ++d) {
      unsigned lds = (unsigned)(size_t)&stage[d * TPB + tx];
      async_copy_b128(lds, x4 + tid + (size_t)d * threads);
    }
    int t = 0;
    // steady state: wait for tile t, consume from LDS, refill slot with tile t+DEPTH
    for (; t + DEPTH < iters; ++t) {
      asm volatile("s_wait_asynccnt 0x7" ::: "memory");      // tile t landed (in-order done)
      const int slot = (t & (DEPTH - 1)) * TPB + tx;
      float4 v = stage[slot];                                // ds_load_b128
      lmin = fminf(lmin, fminf(fminf(v.x, v.y), fminf(v.z, v.w)));
      lmax = fmaxf(lmax, fmaxf(fmaxf(v.x, v.y), fmaxf(v.z, v.w)));
      asm volatile("s_wait_dscnt 0x0" ::: "memory");         // drain read before overwrite
      unsigned lds = (unsigned)(size_t)&stage[slot];
      async_copy_b128(lds, x4 + tid + (size_t)(t + DEPTH) * threads);
    }
    // epilogue: everything issued; drain and consume the last DEPTH tiles
    asm volatile("s_wait_asynccnt 0x0" ::: "memory");
    for (; t < iters; ++t) {
      float4 v = stage[(t & (DEPTH - 1)) * TPB + tx];
      lmin = fminf(lmin, fminf(fminf(v.x, v.y), fminf(v.z, v.w)));
      lmax = fmaxf(lmax, fmaxf(fmaxf(v.x, v.y), fmaxf(v.z, v.w)));
    }
  }
  // remainder (and fallback for tiny n): direct b128 loads
  for (int i = rem_start + tid; i < n4; i += threads) {
    float4 v = x4[i];
    lmin = fminf(lmin, fminf(fminf(v.x, v.y), fminf(v.z, v.w)));
    lmax = fmaxf(lmax, fmaxf(fmaxf(v.x, v.y), fmaxf(v.z, v.w)));
  }
  // scalar tail (n % 4)
  if (blockIdx.x == 0 && threadIdx.x < (n & 3)) {
    float v = x[(n4 << 2) + threadIdx.x];
    lmin = fminf(lmin, v);
    lmax = fmaxf(lmax, v);
  }

  __shared__ float smin[TPB / 32], smax[TPB / 32];
  lmin = wave_min(lmin);
  lmax = wave_max(lmax);
  if ((threadIdx.x & 31) == 0) {
    smin[threadIdx.x >> 5] = lmin;
    smax[threadIdx.x >> 5] = lmax;
  }
  __syncthreads();
  if (threadIdx.x == 0) {
    float m = smin[0], M = smax[0];
#pragma unroll
    for (int i = 1; i < TPB / 32; ++i) { m = fminf(m, smin[i]); M = fmaxf(M, smax[i]); }
    pmin[blockIdx.x] = m;
    pmax[blockIdx.x] = M;
  }
}

// ---- kernel 2 (1 block): final reduce + scalar math + cumsum of old histogram ----
__global__ void hobs_scalars_kernel(const float* __restrict__ hist,
                                    const float* __restrict__ minv,
                                    const float* __restrict__ maxv,
                                    float* __restrict__ ws) {
  __shared__ float smin[TPB / 32], smax[TPB / 32];
  __shared__ float tsum[TPB];
  __shared__ float excl[TPB];

  const float* pmin = ws + OFF_PMIN;
  const float* pmax = ws + OFF_PMAX;
  const int t = threadIdx.x;

  // reduce per-block partials
  float lmin = INFINITY, lmax = -INFINITY;
  for (int i = t; i < NB_MM; i += TPB) {
    lmin = fminf(lmin, pmin[i]);
    lmax = fmaxf(lmax, pmax[i]);
  }
  lmin = wave_min(lmin);
  lmax = wave_max(lmax);
  if ((t & 31) == 0) { smin[t >> 5] = lmin; smax[t >> 5] = lmax; }

  // blocked prefix sum of old histogram: each thread owns 8 consecutive bins
  float loc[8];
  float s = 0.f;
#pragma unroll
  for (int k = 0; k < 8; ++k) { s += hist[t * 8 + k]; loc[k] = s; }
  tsum[t] = s;
  __syncthreads();

  if (t == 0) {
    float m = smin[0], M = smax[0];
#pragma unroll
    for (int i = 1; i < TPB / 32; ++i) { m = fminf(m, smin[i]); M = fmaxf(M, smax[i]); }
    const float minval = minv[0];
    const float maxval = maxv[0];
    // mirror the reference's f32 op order exactly
    const float cmin  = fminf(m, minval);
    const float cmax0 = fmaxf(M, maxval);
    const float hbw   = (maxval - minval) / (float)(BINS * UPR);
    const float denom = (float)BINS * hbw;
    const int   dr    = (int)ceilf((cmax0 - cmin) / denom);
    const float e     = (float)dr * denom - (cmax0 - cmin);
    const float cmax  = cmax0 + e;
    const int start_idx = (int)rintf((minval - cmin) / hbw);   // round-half-even == jnp.round
    const float bw    = (cmax - cmin) / (float)BINS;
    ws[OFF_SCAL + 0] = cmin;
    ws[OFF_SCAL + 1] = bw;
    ((int*)ws)[OFF_SCAL + 2] = start_idx;
    ((int*)ws)[OFF_SCAL + 3] = dr;
    // exclusive scan over thread totals (256 serial adds, negligible)
    float run = 0.f;
    for (int i = 0; i < TPB; ++i) { excl[i] = run; run += tsum[i]; }
  }
  __syncthreads();

  float* C = ws + OFF_C;
  const float p = excl[t];
  if (t == 0) C[0] = 0.f;
#pragma unroll
  for (int k = 0; k < 8; ++k) C[t * 8 + k + 1] = p + loc[k];
}

// ---- kernel 3: histc via 4-way bank-skewed LDS histograms + global u32 atomics ----
__global__ void hobs_hist_kernel(const float* __restrict__ x, int n,
                                 const float* __restrict__ ws,
                                 unsigned* __restrict__ counts) {
  __shared__ unsigned sh[NC * CSTRIDE];
  for (int i = threadIdx.x; i < NC * CSTRIDE; i += TPB) sh[i] = 0u;

  const float cmin = ws[OFF_SCAL + 0];
  const float bw   = ws[OFF_SCAL + 1];
  __syncthreads();

  const float4* __restrict__ x4 = (const float4*)x;
  const int n4 = n >> 2;
  const int stride = gridDim.x * blockDim.x;
  // reversed block order: tail of pass 1 is still resident in the 192MB L2
  const int bid = gridDim.x - 1 - blockIdx.x;
  const unsigned base = (threadIdx.x & (NC - 1)) * CSTRIDE;

  for (int i = bid * blockDim.x + threadIdx.x; i < n4; i += stride) {
    __builtin_prefetch((const void*)(x4 + i + stride), 0, 0);   // global_prefetch_b8
    float4 v = x4[i];
    // exact IEEE division matches jnp bin assignment bit-for-bit
    int i0 = (int)floorf((v.x - cmin) / bw);
    int i1 = (int)floorf((v.y - cmin) / bw);
    int i2 = (int)floorf((v.z - cmin) / bw);
    int i3 = (int)floorf((v.w - cmin) / bw);
    i0 = min(max(i0, 0), BINS - 1);
    i1 = min(max(i1, 0), BINS - 1);
    i2 = min(max(i2, 0), BINS - 1);
    i3 = min(max(i3, 0), BINS - 1);
    atomicAdd(&sh[base + i0], 1u);   // ds_add_u32
    atomicAdd(&sh[base + i1], 1u);
    atomicAdd(&sh[base + i2], 1u);
    atomicAdd(&sh[base + i3], 1u);
  }
  if (blockIdx.x == 0 && threadIdx.x < (n & 3)) {
    float v = x[(n4 << 2) + threadIdx.x];
    int i0 = min(max((int)floorf((v - cmin) / bw), 0), BINS - 1);
    atomicAdd(&sh[base + i0], 1u);
  }
  __syncthreads();

  for (int b = threadIdx.x; b < BINS; b += TPB) {
    unsigned s = 0;
#pragma unroll
    for (int c = 0; c < NC; ++c) s += sh[c * CSTRIDE + b];
    if (s) atomicAdd(&counts[b], s);   // global_atomic_add_u32
  }
}

// ---- kernel 4: merged[j] = new_hist[j] + (F((j+1)dr) - F(j dr)) / UP ----
__global__ void hobs_merge_kernel(const float* __restrict__ hist,
                                  const float* __restrict__ ws,
                                  const unsigned* __restrict__ counts,
                                  float* __restrict__ out) {
  const int j = blockIdx.x * blockDim.x + threadIdx.x;
  if (j >= BINS) return;
  const int start_idx = ((const int*)ws)[OFF_SCAL + 2];
  const int dr        = ((const int*)ws)[OFF_SCAL + 3];
  const float* C = ws + OFF_C;

  auto F = [&](int q) -> float {
    int qc = q - start_idx;
    qc = qc < 0 ? 0 : (qc > BINS * UPR ? BINS * UPR : qc);
    const int k = qc >> 7;                       // qc / UP
    const float r = (float)(qc - (k << 7));
    const int kh = k < BINS - 1 ? k : BINS - 1;
    return (float)UPR * C[k] + r * hist[kh];
  };

  const float interp = (F((j + 1) * dr) - F(j * dr)) * (1.0f / (float)UPR);
  out[j] = (float)counts[j] + interp;
}

extern "C" void kernel_launch(void* const* d_in, const int* in_sizes, int n_in,
                              void* d_out, int out_size, void* d_ws, size_t ws_size,
                              hipStream_t stream) {
  const float* x    = (const float*)d_in[0];
  const float* hist = (const float*)d_in[1];
  const float* minv = (const float*)d_in[2];
  const float* maxv = (const float*)d_in[3];
  const int n = in_sizes[0];

  float*    ws     = (float*)d_ws;
  unsigned* counts = (unsigned*)(ws + OFF_COUNTS);
  float*    out    = (float*)d_out;

  hobs_init_kernel<<<(BINS + TPB - 1) / TPB, TPB, 0, stream>>>(counts);
  hobs_minmax_kernel<<<NB_MM, TPB, 0, stream>>>(x, n, ws + OFF_PMIN, ws + OFF_PMAX);
  hobs_scalars_kernel<<<1, TPB, 0, stream>>>(hist, minv, maxv, ws);
  hobs_hist_kernel<<<NB_H, TPB, 0, stream>>>(x, n, ws, counts);
  hobs_merge_kernel<<<BINS / TPB, TPB, 0, stream>>>(hist, ws, counts, out);
}